// UpstreamNetwork_66726611911213
// MI455X (gfx1250) — compile-verified
//
#include <hip/hip_runtime.h>
#include <hip/hip_bf16.h>

typedef __attribute__((ext_vector_type(2))) float v2f;
typedef __attribute__((ext_vector_type(8))) float v8f;

// Problem constants (from reference)
enum : int {
    VOCAB  = 32000,
    EMB    = 2048,
    NCOMP  = 64,
    SEQL   = 128,
    BATCH  = 4096,
};

// ---------------------------------------------------------------------------
// Stage 1: gather + mean-pool, writing pooled TRANSPOSED: poolT[d*64 + n]
// grid = (NCOMP, EMB/256), block = 256 threads.
// Token ids are uniform across the block -> scalar loads; embedding reads are
// fully coalesced 1024B per row-chunk per block.
// ---------------------------------------------------------------------------
__global__ __launch_bounds__(256)
void pool_mean_kernel(const int* __restrict__ ids,
                      const float* __restrict__ emb,
                      float* __restrict__ poolT) {
    const int n = blockIdx.x;                       // component 0..63
    const int d = blockIdx.y * 256 + threadIdx.x;   // embedding column
    const int* __restrict__ row_ids = ids + n * SEQL;

    float acc = 0.0f;
#pragma unroll 4
    for (int t = 0; t < SEQL; ++t) {
        const int id = row_ids[t];                  // wave-uniform -> s_load
        acc += emb[(size_t)id * EMB + d];
    }
    poolT[(size_t)d * NCOMP + n] = acc * (1.0f / (float)SEQL);
}

// ---------------------------------------------------------------------------
// Stage 2: out[4096,2048] = ratio[4096,64] @ pooled[64,2048]
// One 16x16 tile per wave via V_WMMA_F32_16X16X4_F32, K=64 -> 16 WMMA steps.
// Tiles: 256 (M) x 128 (N) = 32768; 8 waves/block -> 4096 blocks, all full
// (EXEC must be all-1s for WMMA).
//
// A fragment (16x4 f32): lanes 0-15 carry M=lane, K={kb,kb+1} in v0/v1;
// lanes 16-31 carry M=lane-16, K={kb+2,kb+3}. -> float2 at ratio[row*64+kb+2*hi]
// B fragment (4x16 f32): symmetric with N indexed by lane; poolT is [N=2048,K=64]
// row-major, so the same contiguous float2 pattern applies.
// C/D (16x16 f32, 8 VGPRs): VGPR r = row r (lanes 0-15) / row r+8 (lanes 16-31).
// ---------------------------------------------------------------------------
__global__ __launch_bounds__(256)
void mix_wmma_kernel(const float* __restrict__ ratio,
                     const float* __restrict__ poolT,
                     float* __restrict__ out) {
    const int lane = threadIdx.x & 31;
    const int wave = threadIdx.x >> 5;
    const int tile = blockIdx.x * 8 + wave;         // 0..32767
    const int tM = tile >> 7;                       // 0..255
    const int tN = tile & 127;                      // 0..127

    const int l15 = lane & 15;
    const int hi  = lane >> 4;                      // 0 or 1

    const float* __restrict__ aptr = ratio + ((size_t)(tM * 16 + l15) * NCOMP) + 2 * hi;
    const float* __restrict__ bptr = poolT + ((size_t)(tN * 16 + l15) * NCOMP) + 2 * hi;

    v8f acc = {0.f, 0.f, 0.f, 0.f, 0.f, 0.f, 0.f, 0.f};

#pragma unroll
    for (int kk = 0; kk < 16; ++kk) {               // K = 64 = 16 * 4
        v2f a = *(const v2f*)(aptr + kk * 4);
        v2f b = *(const v2f*)(bptr + kk * 4);
        // 8 args: (neg_a, A, neg_b, B, c_mod, C, reuse_a, reuse_b)
        acc = __builtin_amdgcn_wmma_f32_16x16x4_f32(
            false, a, false, b, (short)0, acc, false, false);
    }

    // Store D per the 16x16 f32 C/D layout.
    float* __restrict__ o = out + (size_t)(tM * 16 + hi * 8) * EMB + tN * 16 + l15;
#pragma unroll
    for (int r = 0; r < 8; ++r) {
        o[(size_t)r * EMB] = acc[r];
    }
}

// ---------------------------------------------------------------------------
// Launch: d_in[0]=input_ids (int[64*128]), d_in[1]=input_ratio (f32[4096*64]),
// d_in[2]=emb_weight (f32[32000*2048]); d_out = f32[4096*2048].
// d_ws holds poolT: 2048*64 f32 = 512 KB.
// ---------------------------------------------------------------------------
extern "C" void kernel_launch(void* const* d_in, const int* in_sizes, int n_in,
                              void* d_out, int out_size, void* d_ws, size_t ws_size,
                              hipStream_t stream) {
    const int*   ids   = (const int*)d_in[0];
    const float* ratio = (const float*)d_in[1];
    const float* emb   = (const float*)d_in[2];
    float*       out   = (float*)d_out;
    float*       poolT = (float*)d_ws;              // [EMB, NCOMP]

    dim3 g1(NCOMP, EMB / 256);
    pool_mean_kernel<<<g1, 256, 0, stream>>>(ids, emb, poolT);

    const int tiles = (BATCH / 16) * (EMB / 16);    // 32768
    mix_wmma_kernel<<<tiles / 8, 256, 0, stream>>>(ratio, poolT, out);
}